// POTSlicedWassersteinDistance_83958020702548
// MI455X (gfx1250) — compile-verified
//
#include <hip/hip_runtime.h>
#include <math.h>

typedef __attribute__((ext_vector_type(2))) float v2f;
typedef __attribute__((ext_vector_type(8))) float v8f;

#define B_SZ 64
#define T_SZ 1024
#define D_SZ 256
#define K_SZ 128
#define LDS_STRIDE 260   // 256 + 4 pad floats -> conflict-free across 64 banks

__device__ __forceinline__ float sanitize_f(float v) {
  if (__builtin_isnan(v)) return 0.0f;
  if (v == INFINITY) return 1e4f;
  if (v == -INFINITY) return -1e4f;
  return v;
}

// ---------------------------------------------------------------------------
// Kernel 1: per-token inverse L2 norm (with sanitize), one wave32 per token.
// ---------------------------------------------------------------------------
__global__ __launch_bounds__(256) void swd_inv_norm(
    const float* __restrict__ src, float* __restrict__ invn, int ntok) {
  const int wave = (blockIdx.x * blockDim.x + threadIdx.x) >> 5;
  const int lane = threadIdx.x & 31;
  if (wave >= ntok) return;
  const float* p = src + (size_t)wave * D_SZ;
  float s = 0.0f;
#pragma unroll
  for (int i = 0; i < D_SZ / 32; ++i) {
    float v = sanitize_f(p[lane + i * 32]);
    s += v * v;
  }
#pragma unroll
  for (int m = 16; m > 0; m >>= 1) s += __shfl_xor(s, m, 32);
  if (lane == 0) invn[wave] = 1.0f / fmaxf(sqrtf(s), 1e-6f);
}

// ---------------------------------------------------------------------------
// Kernel 2: fused sanitize/scale + projection GEMM via V_WMMA_F32_16X16X4_F32.
// Block = 256 threads = 8 waves. Block covers 16 tokens x all 128 projections
// for one batch; each wave owns one 16(k) x 16(t) output tile.
// A = proj^T tile (M=k, Kdim=d), B = normalized token tile from LDS (N=t).
// C layout (32-bit 16x16): lane l, vgpr r -> M = r + 8*(l>=16), N = l&15,
// so stores are T-contiguous per half-wave into the (B,K,T) buffer.
// ---------------------------------------------------------------------------
__global__ __launch_bounds__(256) void swd_proj_wmma(
    const float* __restrict__ src, const float* __restrict__ invn,
    const float* __restrict__ proj, float* __restrict__ dst) {
  __shared__ float lds[16 * LDS_STRIDE];
  const int t0  = blockIdx.x * 16;
  const int b   = blockIdx.y;
  const int tid = threadIdx.x;

  // Stage 16 tokens x 256 features, sanitized and scaled by 1/||x||.
  for (int e = tid; e < 16 * D_SZ; e += 256) {
    const int tok = e >> 8;      // /256
    const int d   = e & (D_SZ - 1);
    float v = sanitize_f(src[((size_t)(b * T_SZ + t0 + tok)) * D_SZ + d]);
    lds[tok * LDS_STRIDE + d] = v * invn[b * T_SZ + t0 + tok];
  }
  __syncthreads();

  const int wave  = tid >> 5;         // 0..7  -> k-tile
  const int lane  = tid & 31;
  const int k0    = wave * 16;
  const int half  = lane >> 4;        // 0: Kdim {0,1}, 1: Kdim {2,3}
  const int mn    = lane & 15;        // M for A (k), N for B (t)
  const int dbase = half * 2;

  const float* ap = proj + k0 + mn;          // proj is (D,K) row-major
  const float* bp = &lds[mn * LDS_STRIDE];

  v8f c = {};
#pragma unroll 8
  for (int d0 = 0; d0 < D_SZ; d0 += 4) {
    v2f a, bb;
    a.x  = ap[(size_t)(d0 + dbase) * K_SZ];
    a.y  = ap[(size_t)(d0 + dbase + 1) * K_SZ];
    bb.x = bp[d0 + dbase];
    bb.y = bp[d0 + dbase + 1];
    c = __builtin_amdgcn_wmma_f32_16x16x4_f32(
        /*neg_a=*/false, a, /*neg_b=*/false, bb,
        /*c_mod=*/(short)0, c, /*reuse_a=*/false, /*reuse_b=*/false);
  }

  float* o = dst + ((size_t)(b * K_SZ + k0 + 8 * half)) * T_SZ + t0 + mn;
#pragma unroll
  for (int r = 0; r < 8; ++r) o[(size_t)r * T_SZ] = c[r];
}

// ---------------------------------------------------------------------------
// Kernel 3: per (b,k) row -> bitonic sort x-row and y-row (1024 each) in LDS,
// then sum of squared differences. 512 threads, each handles 2 elements.
// ---------------------------------------------------------------------------
__global__ __launch_bounds__(512) void swd_sort_cost(
    const float* __restrict__ xp, const float* __restrict__ yp,
    float* __restrict__ cost) {
  __shared__ float sx[T_SZ];
  __shared__ float sy[T_SZ];
  const int row = blockIdx.x;          // b*K + k
  const int tid = threadIdx.x;         // 0..511

  const float* px = xp + (size_t)row * T_SZ;
  const float* py = yp + (size_t)row * T_SZ;
  sx[tid] = px[tid];            sx[tid + 512] = px[tid + 512];
  sy[tid] = py[tid];            sy[tid + 512] = py[tid + 512];
  __syncthreads();

  for (int k = 2; k <= T_SZ; k <<= 1) {
    for (int j = k >> 1; j > 0; j >>= 1) {
#pragma unroll
      for (int w = 0; w < 2; ++w) {
        const int i   = tid + w * 512;
        const int ixj = i ^ j;
        if (ixj > i) {
          const bool up = ((i & k) == 0);
          float a = sx[i], b = sx[ixj];
          if ((a > b) == up) { sx[i] = b; sx[ixj] = a; }
          float c = sy[i], d = sy[ixj];
          if ((c > d) == up) { sy[i] = d; sy[ixj] = c; }
        }
      }
      __syncthreads();
    }
  }

  float s = 0.0f;
#pragma unroll
  for (int w = 0; w < 2; ++w) {
    const int i = tid + w * 512;
    const float d = sx[i] - sy[i];
    s += d * d;                        // P = 2
  }
  __syncthreads();
  sx[tid] = s;
  __syncthreads();
  for (int stride = 256; stride > 0; stride >>= 1) {
    if (tid < stride) sx[tid] += sx[tid + stride];
    __syncthreads();
  }
  if (tid == 0) cost[row] = sx[0];     // raw sum over T; scaled in finalize
}

// ---------------------------------------------------------------------------
// Kernel 4: per batch, mean over K and T, then sqrt + output sanitize.
// ---------------------------------------------------------------------------
__global__ __launch_bounds__(128) void swd_finalize(
    const float* __restrict__ cost, float* __restrict__ out) {
  __shared__ float s[K_SZ];
  const int b   = blockIdx.x;
  const int tid = threadIdx.x;         // 0..127
  s[tid] = cost[b * K_SZ + tid];
  __syncthreads();
  for (int stride = 64; stride > 0; stride >>= 1) {
    if (tid < stride) s[tid] += s[tid + stride];
    __syncthreads();
  }
  if (tid == 0) {
    float v = sqrtf(s[0] / (float)(K_SZ * T_SZ));
    if (__builtin_isnan(v)) v = 1e4f;
    else if (v == INFINITY) v = 1e4f;
    else if (v == -INFINITY) v = 0.0f;
    out[b] = v;
  }
}

// ---------------------------------------------------------------------------
extern "C" void kernel_launch(void* const* d_in, const int* in_sizes, int n_in,
                              void* d_out, int out_size, void* d_ws, size_t ws_size,
                              hipStream_t stream) {
  (void)in_sizes; (void)n_in; (void)out_size; (void)ws_size;
  const float* x    = (const float*)d_in[0];   // (B,T,D)
  const float* y    = (const float*)d_in[1];   // (B,T,D)
  const float* proj = (const float*)d_in[2];   // (D,K)
  float* out = (float*)d_out;                  // (B,)

  const int NTOK = B_SZ * T_SZ;                // 65536 tokens per tensor
  float* ws   = (float*)d_ws;
  float* invx = ws;                            // NTOK
  float* invy = invx + NTOK;                   // NTOK
  float* xp   = invy + NTOK;                   // B*K*T
  float* yp   = xp + (size_t)B_SZ * K_SZ * T_SZ;
  float* cost = yp + (size_t)B_SZ * K_SZ * T_SZ;   // B*K

  // 1) inverse norms (one wave32 per token)
  {
    const int waves_per_block = 256 / 32;
    const int blocks = (NTOK + waves_per_block - 1) / waves_per_block;
    swd_inv_norm<<<blocks, 256, 0, stream>>>(x, invx, NTOK);
    swd_inv_norm<<<blocks, 256, 0, stream>>>(y, invy, NTOK);
  }

  // 2) fused sanitize/normalize + projection GEMM (fp32 WMMA)
  {
    dim3 grid(T_SZ / 16, B_SZ);
    swd_proj_wmma<<<grid, 256, 0, stream>>>(x, invx, proj, xp);
    swd_proj_wmma<<<grid, 256, 0, stream>>>(y, invy, proj, yp);
  }

  // 3) paired bitonic sort + squared-diff cost per (b,k) row
  swd_sort_cost<<<B_SZ * K_SZ, 512, 0, stream>>>(xp, yp, cost);

  // 4) reduce over K, scale, sqrt, sanitize
  swd_finalize<<<B_SZ, 128, 0, stream>>>(cost, out);
}